// CRF_67010079752423
// MI455X (gfx1250) — compile-verified
//
#include <hip/hip_runtime.h>

#define B_ 16384
#define T_ 128
#define L_ 9

constexpr float LOG2E = 1.4426950408889634f;
constexpr float LN2f  = 0.6931471805599453f;

typedef __attribute__((ext_vector_type(2))) float v2f;
typedef __attribute__((ext_vector_type(8))) float v8f;

// half-wave swap (lane ^= 16) as a VALU op: v_permlanex16_b32 with identity
// lane selects pulls from the opposite 16-lane group at the same position.
__device__ __forceinline__ float swap16(float x) {
  int xi = __float_as_int(x);
  int v  = __builtin_amdgcn_permlanex16(xi, xi, 0x76543210, 0xfedcba98, false, false);
  return __int_as_float(v);
}

// ---------------------------------------------------------------------------
// Kernel 1: CRF forward algorithm in exp space, 16 sequences per wave.
//   v'_j = sum_i exp(trans[i][j]) * v_i     -> 3x V_WMMA_F32_16X16X4_F32
//   v'_j *= exp2(logit * log2e)             -> 1 exp per logit element
//   renormalize each step; accumulate log2 scale in c2.
// Writes P[seq] = log_norm (natural log).
// ---------------------------------------------------------------------------
__global__ __launch_bounds__(256) void crf_forward(
    const float* __restrict__ logits, const int* __restrict__ seq_lens,
    const float* __restrict__ trans, float* __restrict__ P) {
  const int lane = threadIdx.x & 31;
  const int wave = (blockIdx.x * blockDim.x + threadIdx.x) >> 5;
  const int s16  = lane & 15;          // sequence slot (column N)
  const int hi   = lane >> 4;          // upper half-wave flag
  const int seq  = wave * 16 + s16;
  const float* base = logits + (size_t)seq * (T_ * L_);
  const int len = seq_lens[seq];

  // A = Mexp^T in 16x4 f32 A-layout, 3 K-chunks. Rows/cols >= 9 are zero, so
  // D rows 9..15 are EXACTLY 0 every step (exploited below: no lane selects
  // needed for the per-sequence state sum).
  float a[3][2];
  #pragma unroll
  for (int c = 0; c < 3; ++c) {
    #pragma unroll
    for (int p = 0; p < 2; ++p) {
      int i = 4 * c + p + 2 * hi;      // input state (K)
      int j = s16;                     // output state (M = lane%16)
      a[c][p] = (i < L_ && j < L_)
                    ? __builtin_amdgcn_exp2f(trans[i * L_ + j] * LOG2E)
                    : 0.0f;
    }
  }

  // B-layout init from t=0 logits. Chunk c VGPR p holds rows {4c+p, 4c+p+2}.
  float b0 = __builtin_amdgcn_exp2f(base[hi ? 2 : 0] * LOG2E);
  float b1 = __builtin_amdgcn_exp2f(base[hi ? 3 : 1] * LOG2E);
  float b2 = __builtin_amdgcn_exp2f(base[hi ? 6 : 4] * LOG2E);
  float b3 = __builtin_amdgcn_exp2f(base[hi ? 7 : 5] * LOG2E);
  float b4 = hi ? 0.0f : __builtin_amdgcn_exp2f(base[8] * LOG2E);
  float b5 = 0.0f;
  float c2 = 0.0f;                     // accumulated log2 of scales

  for (int t = 1; t < T_; ++t) {
    const bool mt = (t < len);
    // all sequences in this wave finished -> skip remaining steps (uniform
    // branch: EXEC stays all-ones as WMMA requires)
    if (__builtin_amdgcn_ballot_w32(mt) == 0u) break;

    const float* lp = base + t * L_;
    __builtin_prefetch(lp + 4 * L_, 0, 3);

    float lg[9];
    #pragma unroll
    for (int r = 0; r < 9; ++r) lg[r] = lp[r];

    v2f A0 = {a[0][0], a[0][1]}, A1 = {a[1][0], a[1][1]}, A2 = {a[2][0], a[2][1]};
    v2f B0 = {b0, b1}, B1 = {b2, b3}, B2 = {b4, b5};
    v8f d = {};
    d = __builtin_amdgcn_wmma_f32_16x16x4_f32(false, A0, false, B0, (short)0, d, false, false);
    d = __builtin_amdgcn_wmma_f32_16x16x4_f32(false, A1, false, B1, (short)0, d, false, false);
    d = __builtin_amdgcn_wmma_f32_16x16x4_f32(false, A2, false, B2, (short)0, d, false, false);

    // multiply by exp(logit). D VGPR r holds rows {r, r+8}; hi half of d[0] is
    // row 8, hi halves of d[1..7] are rows 9..15 == 0.
    float e[8];
    e[0] = d[0] * __builtin_amdgcn_exp2f((hi ? lg[8] : lg[0]) * LOG2E);
    #pragma unroll
    for (int r = 1; r < 8; ++r)
      e[r] = d[r] * __builtin_amdgcn_exp2f(lg[r] * LOG2E);

    // per-sequence total over states 0..8.  Unconditional sum: on hi lanes
    // e[1..7]==0 so w == e[0] (row 8) automatically -> no divergent select.
    float w = ((e[0] + e[1]) + (e[2] + e[3])) + ((e[4] + e[5]) + (e[6] + e[7]));
    float tot   = w + swap16(w);
    float dc    = __builtin_amdgcn_logf(tot);   // log2
    float scale = __builtin_amdgcn_rcpf(tot);

    #pragma unroll
    for (int r = 0; r < 8; ++r) e[r] *= scale;

    // D-layout -> next-step B-layout (cross-half moves on full EXEC)
    float s2 = swap16(e[2]);
    float s3 = swap16(e[3]);
    float s6 = swap16(e[6]);
    float s7 = swap16(e[7]);
    float s0 = swap16(e[0]);           // row 8 into lower half
    float nb0 = hi ? s2 : e[0];        // rows 0 | 2
    float nb1 = hi ? s3 : e[1];        // rows 1 | 3
    float nb2 = hi ? s6 : e[4];        // rows 4 | 6
    float nb3 = hi ? s7 : e[5];        // rows 5 | 7
    float nb4 = hi ? 0.0f : s0;        // rows 8 | (10 = 0)

    // masked step: sequences past their length keep alphas unchanged
    b0 = mt ? nb0 : b0;
    b1 = mt ? nb1 : b1;
    b2 = mt ? nb2 : b2;
    b3 = mt ? nb3 : b3;
    b4 = mt ? nb4 : b4;
    c2 = mt ? (c2 + dc) : c2;
  }

  // final logsumexp over states 0..8: lo lanes hold rows {0,1,4,5,8},
  // hi lanes rows {2,3,6,7}; cross-half add completes the sum in all lanes.
  float z   = (((b0 + b1) + (b2 + b3)) + b4);
  float tot = z + swap16(z);
  if (!hi) P[seq] = (c2 + __builtin_amdgcn_logf(tot)) * LN2f;  // natural log_norm
}

// ---------------------------------------------------------------------------
// Kernel 2: unary + transition sequence score (gathers hit L2: logits 75.5MB
// < 192MB L2, just streamed by kernel 1).  P[s] -= seq_score.
// ---------------------------------------------------------------------------
__global__ __launch_bounds__(256) void crf_score(
    const float* __restrict__ logits, const int* __restrict__ labels,
    const int* __restrict__ seq_lens, const float* __restrict__ trans,
    float* __restrict__ P) {
  __shared__ float tr[L_ * L_];
  if (threadIdx.x < L_ * L_) tr[threadIdx.x] = trans[threadIdx.x];
  __syncthreads();

  const int s = blockIdx.x * blockDim.x + threadIdx.x;
  if (s >= B_) return;
  const int*   lab = labels + s * T_;
  const float* lg  = logits + (size_t)s * (T_ * L_);
  const int len = seq_lens[s];

  int prev = lab[0];
  float acc = lg[prev];                          // t=0 unary (len >= 1)
  for (int t = 1; t < len; ++t) {
    int l = lab[t];
    acc += lg[t * L_ + l] + tr[prev * L_ + l];   // unary + transition, mask t<len
    prev = l;
  }
  P[s] -= acc;                                   // P[s] = log_norm - seq_score
}

// ---------------------------------------------------------------------------
// Kernel 3: deterministic tree reduction; output = sum(log_norm - score) = -sum(ll)
// ---------------------------------------------------------------------------
__global__ __launch_bounds__(256) void crf_reduce(const float* __restrict__ P,
                                                  float* __restrict__ out) {
  __shared__ float sm[256];
  float acc = 0.0f;
  for (int i = threadIdx.x; i < B_; i += 256) acc += P[i];
  sm[threadIdx.x] = acc;
  __syncthreads();
  for (int off = 128; off > 0; off >>= 1) {
    if ((int)threadIdx.x < off) sm[threadIdx.x] += sm[threadIdx.x + off];
    __syncthreads();
  }
  if (threadIdx.x == 0) out[0] = sm[0];
}

extern "C" void kernel_launch(void* const* d_in, const int* in_sizes, int n_in,
                              void* d_out, int out_size, void* d_ws, size_t ws_size,
                              hipStream_t stream) {
  const float* logits   = (const float*)d_in[0];
  const int*   labels   = (const int*)d_in[1];
  const int*   seq_lens = (const int*)d_in[2];
  const float* trans    = (const float*)d_in[3];
  float* out = (float*)d_out;
  float* P   = (float*)d_ws;           // B_ floats = 64 KB scratch

  crf_forward<<<B_ / 128, 256, 0, stream>>>(logits, seq_lens, trans, P); // 8 waves * 16 seq
  crf_score  <<<B_ / 256, 256, 0, stream>>>(logits, labels, seq_lens, trans, P);
  crf_reduce <<<1, 256, 0, stream>>>(P, out);
}